// SchNETBondEmbedding_41755672052433
// MI455X (gfx1250) — compile-verified
//
#include <hip/hip_runtime.h>
#include <hip/hip_bf16.h>
#include <cstdint>
#include <cstddef>

// Problem constants (match reference)
#define NN     20000
#define EE     320000
#define ZMAX   16
#define DIM    64
#define SUB    16
#define NRBF   8
#define NB     5
#define FCIN   656          // NRBF*SUB*NB + SUB = 640 + 16
#define HDIM   64
#define WPB    8            // waves per block in the FC kernel
#define LOG2F_ 0.6931471805599453f

typedef float v2f __attribute__((ext_vector_type(2)));
typedef float v8f __attribute__((ext_vector_type(8)));
typedef int   i32x4 __attribute__((vector_size(16)));

#if __has_builtin(__builtin_amdgcn_global_load_async_to_lds_b128) && \
    __has_builtin(__builtin_amdgcn_s_wait_asynccnt)
#define USE_ASYNC_LDS 1
#endif

// 16-byte global -> LDS copy (async DMA when available, tracked by ASYNCcnt).
// OFF is a template parameter so it stays an integer constant expression for
// the builtin's immediate-offset operand.
template <int OFF>
__device__ __forceinline__ void cp16(const float* g, float* l) {
#ifdef USE_ASYNC_LDS
    __builtin_amdgcn_global_load_async_to_lds_b128(
        (__attribute__((address_space(1))) i32x4*)(float*)g,
        (__attribute__((address_space(3))) i32x4*)l,
        OFF, 0);
#else
    *(float4*)((char*)l + OFF) = *(const float4*)((const char*)g + OFF);
#endif
}
__device__ __forceinline__ void cp64(const float* g, float* l) {
    cp16<0>(g, l); cp16<16>(g, l); cp16<32>(g, l); cp16<48>(g, l);
}
__device__ __forceinline__ void wait_async4() {
#ifdef USE_ASYNC_LDS
    __builtin_amdgcn_s_wait_asynccnt(4);
#endif
}
__device__ __forceinline__ void wait_async0() {
#ifdef USE_ASYNC_LDS
    __builtin_amdgcn_s_wait_asynccnt(0);
#endif
}

__device__ __forceinline__ v8f wmma_f32_4(v2f a, v2f b, v8f c) {
    // fp32 WMMA: D(16x16) = A(16x4) * B(4x16) + C   -> v_wmma_f32_16x16x4_f32
    return __builtin_amdgcn_wmma_f32_16x16x4_f32(
        false, a, false, b, (short)0, c, false, false);
}

__device__ __forceinline__ float ssp(float x) {
    // shifted softplus: log(1+e^x) - log 2, numerically stable
    return fmaxf(x, 0.0f) + log1pf(expf(-fabsf(x))) - LOG2F_;
}

// Accumulate a k-chunk: A rows from `arow` (caller pre-offsets by 2*hi),
// B rows (krows x 64, row-major) from `B` (LDS or global).
__device__ __forceinline__ void gemm_chunk(const float* __restrict__ B,
                                           const float* __restrict__ arow,
                                           int krows, int m16, int hi,
                                           v8f& c0, v8f& c1, v8f& c2, v8f& c3) {
    #pragma unroll 4
    for (int kb = 0; kb < krows; kb += 4) {
        v2f a = *(const v2f*)(arow + kb);
        const float* wA = B + (size_t)(kb + 2 * hi) * HDIM;  // row k
        const float* wB = wA + HDIM;                         // row k+1
        v2f bf;
        bf.x = wA[m16 +  0]; bf.y = wB[m16 +  0]; c0 = wmma_f32_4(a, bf, c0);
        bf.x = wA[m16 + 16]; bf.y = wB[m16 + 16]; c1 = wmma_f32_4(a, bf, c1);
        bf.x = wA[m16 + 32]; bf.y = wB[m16 + 32]; c2 = wmma_f32_4(a, bf, c2);
        bf.x = wA[m16 + 48]; bf.y = wB[m16 + 48]; c3 = wmma_f32_4(a, bf, c3);
    }
}

// -------- CSR row offsets from sorted edge_src (lower_bound) --------
__global__ void rowstart_kernel(const int* __restrict__ edge_src,
                                int* __restrict__ row_start) {
    int n = blockIdx.x * blockDim.x + threadIdx.x;
    if (n > NN) return;
    int lo = 0, hi = EE;
    while (lo < hi) {
        int mid = (lo + hi) >> 1;
        if (edge_src[mid] < n) lo = mid + 1; else hi = mid;
    }
    row_start[n] = lo;
}

// -------- xi = one_hot(species) @ W_species  (row gather) --------
__global__ void embed_kernel(const int* __restrict__ species,
                             const float* __restrict__ Wsp,
                             float* __restrict__ xi) {
    int idx = blockIdx.x * blockDim.x + threadIdx.x;   // n*64 + d
    int n = idx >> 6, d = idx & 63;
    xi[idx] = Wsp[species[n] * DIM + d];
}

// -------- h = xi @ W_sm[l] + b_sm[l] ; si -> dxi tail, mi -> scratch --------
__global__ void sm_kernel(const float* __restrict__ xi,
                          const float* __restrict__ wsm,   // (64,32) for layer l
                          const float* __restrict__ bsm,   // (32,)
                          float* __restrict__ dxi,         // (N,656)
                          float* __restrict__ mi) {        // (N,16)
    int idx = blockIdx.x * blockDim.x + threadIdx.x;       // n*32 + j
    int n = idx >> 5, j = idx & 31;
    const float* x = xi + (size_t)n * DIM;
    const float* w = wsm + j;                              // stride 32 down K
    float acc = bsm[j];
    #pragma unroll 8
    for (int k = 0; k < DIM; ++k) acc = fmaf(x[k], w[k * 32], acc);
    if (j < SUB) dxi[(size_t)n * FCIN + 640 + j] = acc;    // si
    else         mi[(size_t)n * SUB + (j - SUB)] = acc;    // mi
}

// -------- per-node segment-sum of rb ⊗ mi[dst] ⊗ bond_order, scaled by switch ----
__global__ __launch_bounds__(128)
void edge_kernel(const int* __restrict__ row_start,
                 const int* __restrict__ edge_dst,
                 const float* __restrict__ distances,
                 const float* __restrict__ sw_,
                 const float* __restrict__ bo,          // (E,5)
                 const float* __restrict__ mi,          // (N,16)
                 float* __restrict__ dxi) {             // (N,656)
    const int n = blockIdx.x;
    const int t = threadIdx.x;
    __shared__ float s_mi[SUB];
    __shared__ float s_rb[NRBF];
    __shared__ float s_bo[NB];
    __shared__ float s_sw;

    float a0 = 0.f, a1 = 0.f, a2 = 0.f, a3 = 0.f, a4 = 0.f;
    const int e_beg = row_start[n];
    const int e_end = row_start[n + 1];

    for (int e = e_beg; e < e_end; ++e) {
        if (t < 16) {
            int dst = edge_dst[e];
            s_mi[t] = mi[(size_t)dst * SUB + t];
        } else if (t < 24) {
            int r = t - 16;
            float d = distances[e];
            // mu_r = r*CUTOFF/(NRBF-1) = r*5/7 ; sigma = CUTOFF/NRBF = 5/8
            float u = (d - (float)r * (5.0f / 7.0f)) * 1.6f;
            s_rb[r] = expf(-0.5f * u * u);
        } else if (t < 29) {
            s_bo[t - 24] = bo[(size_t)e * NB + (t - 24)];
        } else if (t == 29) {
            s_sw = sw_[e];
        }
        __syncthreads();
        float c = s_sw * s_rb[t >> 4] * s_mi[t & 15];
        a0 = fmaf(c, s_bo[0], a0);
        a1 = fmaf(c, s_bo[1], a1);
        a2 = fmaf(c, s_bo[2], a2);
        a3 = fmaf(c, s_bo[3], a3);
        a4 = fmaf(c, s_bo[4], a4);
        __syncthreads();
    }
    float* o = dxi + (size_t)n * FCIN + t * NB;
    o[0] = a0; o[1] = a1; o[2] = a2; o[3] = a3; o[4] = a4;
}

// -------- fused 3-GEMM MLP; 8 waves/block, weights async-staged into LDS --------
__global__ __launch_bounds__(256)
void fc_fused_kernel(const float* __restrict__ dxi,   // (N,656)
                     const float* __restrict__ w0, const float* __restrict__ b0,
                     const float* __restrict__ w1, const float* __restrict__ b1,
                     const float* __restrict__ w2, const float* __restrict__ b2,
                     float* __restrict__ xi) {        // (N,64) in/out (residual)
    __shared__ float wbuf0[64 * HDIM];                // 16 KB weight buffer A
    __shared__ float wbuf1[64 * HDIM];                // 16 KB weight buffer B
    __shared__ float lds_act[WPB][16 * HDIM];         // per-wave 16x64 activations

    const int tid  = threadIdx.x;
    const int lane = tid & 31;
    const int wave = tid >> 5;
    const int m16  = lane & 15;
    const int hi   = (lane >> 4) & 1;
    const int node0 = blockIdx.x * (WPB * 16) + wave * 16;
    const bool active = (node0 < NN);                 // wave-uniform

    // ---- kick off async fill of w0 chunk 0 (rows 0..63); 64 B per thread ----
    cp64(w0 + tid * 16, wbuf0 + tid * 16);

    // ===== GEMM0: (16 x 656) @ (656 x 64) =====
    v8f c0 = {}, c1 = {}, c2 = {}, c3 = {};
    const float* arow = dxi + (size_t)(node0 + m16) * FCIN + 2 * hi;

    // si tail (rows 640..655) straight from global, overlapping the DMA above
    if (active) gemm_chunk(w0 + 640 * HDIM, arow + 640, 16, m16, hi, c0, c1, c2, c3);

    // 10 double-buffered 64-row chunks of w0
    for (int c = 0; c < 10; ++c) {
        const float* cur = (c & 1) ? wbuf1 : wbuf0;
        if (c < 9) {
            float* nxt = (c & 1) ? wbuf0 : wbuf1;
            cp64(w0 + (size_t)(c + 1) * 64 * HDIM + tid * 16, nxt + tid * 16);
            wait_async4();                            // chunk c landed, c+1 in flight
        } else {
            wait_async0();
        }
        __syncthreads();
        if (active) {
            if (c < 9) __builtin_prefetch(arow + (c + 1) * 64, 0, 1);
            gemm_chunk(cur, arow + c * 64, 64, m16, hi, c0, c1, c2, c3);
        }
        __syncthreads();                              // buffer reusable
    }

    // ---- stage w1 -> wbuf0 and w2 -> wbuf1 (8 async ops/thread, in order) ----
    cp64(w1 + tid * 16, wbuf0 + tid * 16);
    cp64(w2 + tid * 16, wbuf1 + tid * 16);

    // bias + ssp -> per-wave LDS activations (overlaps the weight DMA)
    float* act = lds_act[wave];
    if (active) {
        float bb0 = b0[m16], bb1 = b0[m16 + 16], bb2 = b0[m16 + 32], bb3 = b0[m16 + 48];
        #pragma unroll
        for (int i = 0; i < 8; ++i) {
            int m = i + 8 * hi;                       // D layout: reg i -> row m
            act[m * HDIM + m16 +  0] = ssp(c0[i] + bb0);
            act[m * HDIM + m16 + 16] = ssp(c1[i] + bb1);
            act[m * HDIM + m16 + 32] = ssp(c2[i] + bb2);
            act[m * HDIM + m16 + 48] = ssp(c3[i] + bb3);
        }
    }
    wait_async4();                                    // w1's 4 ops done (w2 may fly)
    __syncthreads();

    // ===== GEMM1: (16 x 64) @ (64 x 64) =====
    v8f d0 = {}, d1 = {}, d2 = {}, d3 = {};
    if (active) {
        gemm_chunk(wbuf0, act + m16 * HDIM + 2 * hi, 64, m16, hi, d0, d1, d2, d3);
        float bb0 = b1[m16], bb1 = b1[m16 + 16], bb2 = b1[m16 + 32], bb3 = b1[m16 + 48];
        #pragma unroll
        for (int i = 0; i < 8; ++i) {                 // per-wave region; DS in-order
            int m = i + 8 * hi;
            act[m * HDIM + m16 +  0] = ssp(d0[i] + bb0);
            act[m * HDIM + m16 + 16] = ssp(d1[i] + bb1);
            act[m * HDIM + m16 + 32] = ssp(d2[i] + bb2);
            act[m * HDIM + m16 + 48] = ssp(d3[i] + bb3);
        }
    }
    wait_async0();                                    // w2 resident
    __syncthreads();

    // ===== GEMM2: (16 x 64) @ (64 x 64), residual into xi =====
    if (active) {
        v8f e0 = {}, e1 = {}, e2 = {}, e3 = {};
        gemm_chunk(wbuf1, act + m16 * HDIM + 2 * hi, 64, m16, hi, e0, e1, e2, e3);
        float bb0 = b2[m16], bb1 = b2[m16 + 16], bb2 = b2[m16 + 32], bb3 = b2[m16 + 48];
        float* xo = xi + (size_t)node0 * DIM;
        #pragma unroll
        for (int i = 0; i < 8; ++i) {
            int m = i + 8 * hi;
            xo[m * DIM + m16 +  0] += e0[i] + bb0;
            xo[m * DIM + m16 + 16] += e1[i] + bb1;
            xo[m * DIM + m16 + 32] += e2[i] + bb2;
            xo[m * DIM + m16 + 48] += e3[i] + bb3;
        }
    }
}

extern "C" void kernel_launch(void* const* d_in, const int* in_sizes, int n_in,
                              void* d_out, int out_size, void* d_ws, size_t ws_size,
                              hipStream_t stream) {
    const int*   species  = (const int*)  d_in[0];
    const int*   edge_src = (const int*)  d_in[1];
    const int*   edge_dst = (const int*)  d_in[2];
    const float* dist     = (const float*)d_in[3];
    const float* sw       = (const float*)d_in[4];
    const float* bo       = (const float*)d_in[5];
    const float* Wsp      = (const float*)d_in[6];
    const float* Wsm      = (const float*)d_in[7];
    const float* bsm      = (const float*)d_in[8];
    const float* fw0      = (const float*)d_in[9];
    const float* fb0      = (const float*)d_in[10];
    const float* fw1      = (const float*)d_in[11];
    const float* fb1      = (const float*)d_in[12];
    const float* fw2      = (const float*)d_in[13];
    const float* fb2      = (const float*)d_in[14];
    float* xi = (float*)d_out;                       // xi lives in d_out across layers

    // Workspace layout (128B aligned):
    //   [0)          row_start : (N+1) int32  (80,004 B -> 80,128)
    //   [80128)      mi        : N*16 f32     (1,280,000 B)
    //   [1,360,128)  dxi       : N*656 f32    (52,480,000 B)
    char* ws = (char*)d_ws;
    int*   row_start = (int*)ws;
    float* mi        = (float*)(ws + 80128);
    float* dxi       = (float*)(ws + 80128 + 1280000);

    rowstart_kernel<<<(NN + 1 + 255) / 256, 256, 0, stream>>>(edge_src, row_start);
    embed_kernel<<<(NN * DIM) / 256, 256, 0, stream>>>(species, Wsp, xi);

    const int fc_blocks = (NN + WPB * 16 - 1) / (WPB * 16);   // 157
    for (int l = 0; l < 3; ++l) {
        sm_kernel<<<(NN * 32) / 256, 256, 0, stream>>>(
            xi, Wsm + l * DIM * 2 * SUB, bsm + l * 2 * SUB, dxi, mi);
        edge_kernel<<<NN, 128, 0, stream>>>(
            row_start, edge_dst, dist, sw, bo, mi, dxi);
        fc_fused_kernel<<<fc_blocks, 256, 0, stream>>>(
            dxi,
            fw0 + (size_t)l * FCIN * HDIM, fb0 + l * HDIM,
            fw1 + (size_t)l * HDIM * HDIM, fb1 + l * HDIM,
            fw2 + (size_t)l * HDIM * HDIM, fb2 + l * HDIM,
            xi);
    }
}